// GraphModel_15994458211317
// MI455X (gfx1250) — compile-verified
//
#include <hip/hip_runtime.h>
#include <math.h>

// ---------------------------------------------------------------------------
// Edge-feature message passing, algebraically restructured:
//   agg[v] = (deg+1)*x[v]@Wa + (scatter(x)+x[v])@Wb + s[v]*we + (deg+1)*b
// so each layer = one L2-resident scatter-add + one small node-level GEMM
// done with V_WMMA_F32_16X16X4_F32 (full fp32 matrix pipe on gfx1250).
// ---------------------------------------------------------------------------

typedef __attribute__((ext_vector_type(2))) float v2f;
typedef __attribute__((ext_vector_type(8))) float v8f;

#define YSTRIDE 128  // shared stride for the per-node aggregate buffer

__global__ void zero_f32(float* __restrict__ p, long long n) {
    long long i = (long long)blockIdx.x * blockDim.x + threadIdx.x;
    long long stride = (long long)gridDim.x * blockDim.x;
    for (; i < n; i += stride) p[i] = 0.0f;
}

// deg (as float) and sum of edge_attr per destination node
__global__ void edge_stats(const int* __restrict__ dst, const float* __restrict__ ea,
                           float* __restrict__ cnt, float* __restrict__ s, int E) {
    int e = blockIdx.x * blockDim.x + threadIdx.x;
    if (e >= E) return;
    int d = dst[e];
    atomicAdd(&cnt[d], 1.0f);
    atomicAdd(&s[d], ea[e]);
}

// y[dst] += feat[src], C channels per edge (one thread per (edge, channel))
template <int LOGC>
__global__ void scatter_feat(const float* __restrict__ feat, const int* __restrict__ src,
                             const int* __restrict__ dst, float* __restrict__ y, int E) {
    constexpr int C = 1 << LOGC;
    long long idx = (long long)blockIdx.x * blockDim.x + threadIdx.x;
    if (idx >= ((long long)E << LOGC)) return;
    int e = (int)(idx >> LOGC);
    int c = (int)(idx & (C - 1));
    atomicAdd(&y[(long long)dst[e] * YSTRIDE + c], feat[(long long)src[e] * C + c]);
}

// Node-level GEMM with fp32 WMMA.
//  z[v] = [ (deg+1)*x[v] | y[v] + x[v] ]  (2*FIN wide, staged in LDS)
//  out[v] = relu( z[v] @ W[0:2*FIN]  +  s[v]*W[2*FIN]  +  (deg+1)*b )
// One wave per 16x16 output tile; FOUT/16 waves per 16-node block.
template <int FIN, int FOUT>
__global__ __launch_bounds__(32 * (FOUT / 16)) void mp_gemm(
    const float* __restrict__ feat, const float* __restrict__ y,
    const float* __restrict__ cnt, const float* __restrict__ s,
    const float* __restrict__ W, const float* __restrict__ bias,
    float* __restrict__ out) {
    constexpr int K2 = 2 * FIN;
    constexpr int LDZ = K2 + 4;  // pad: stride 132/260 spreads the 64 LDS banks
    __shared__ float zsh[16 * LDZ];

    const int nb = blockIdx.x * 16;
    const int tid = threadIdx.x;

    // cooperative stage of the 16-node z panel
    for (int i = tid; i < 16 * FIN; i += blockDim.x) {
        int r = i / FIN, col = i % FIN;
        int node = nb + r;
        float xv = feat[(long long)node * FIN + col];
        float c = cnt[node] + 1.0f;  // +1 for the self loop
        zsh[r * LDZ + col] = c * xv;
        zsh[r * LDZ + FIN + col] = y[(long long)node * YSTRIDE + col] + xv;
    }
    __syncthreads();

    const int lane = tid & 31;
    const int wave = tid >> 5;
    const int n0 = wave * 16;      // output-column tile
    const int lo = lane & 15;      // M (for A) / N (for B,C) index
    const int hi = lane >> 4;      // K-half selector

    v8f acc = {};
    for (int k = 0; k < K2; k += 4) {
        // A 16x4 f32: lanes 0-15 hold K=k..k+1, lanes 16-31 hold K=k+2..k+3
        int ka = k + 2 * hi;
        v2f a, b;
        a.x = zsh[lo * LDZ + ka];
        a.y = zsh[lo * LDZ + ka + 1];
        // B 4x16 f32: row striped across lanes, same K-half split
        b.x = W[(long long)ka * FOUT + n0 + lo];
        b.y = W[(long long)(ka + 1) * FOUT + n0 + lo];
        acc = __builtin_amdgcn_wmma_f32_16x16x4_f32(
            false, a, false, b, (short)0, acc, false, false);
    }

    // C/D layout: VGPR r -> M = r + 8*hi, N = lane&15
    for (int r = 0; r < 8; ++r) {
        int m = r + 8 * hi;
        int node = nb + m;
        int n = n0 + lo;
        float c = cnt[node] + 1.0f;
        float v = acc[r] + s[node] * W[(long long)K2 * FOUT + n] + c * bias[n];
        out[(long long)node * FOUT + n] = fmaxf(v, 0.0f);
    }
}

// in-place log-softmax over 64 columns; one wave32 per row, 2 cols per lane
__global__ void log_softmax64(float* __restrict__ o, int N) {
    int row = blockIdx.x * 8 + (threadIdx.x >> 5);
    if (row >= N) return;
    int lane = threadIdx.x & 31;
    float* p = o + (long long)row * 64;
    float v0 = p[lane], v1 = p[lane + 32];
    float m = fmaxf(v0, v1);
    for (int off = 16; off > 0; off >>= 1) m = fmaxf(m, __shfl_xor(m, off, 32));
    float sum = expf(v0 - m) + expf(v1 - m);
    for (int off = 16; off > 0; off >>= 1) sum += __shfl_xor(sum, off, 32);
    float l = m + logf(sum);
    p[lane] = v0 - l;
    p[lane + 32] = v1 - l;
}

extern "C" void kernel_launch(void* const* d_in, const int* in_sizes, int n_in,
                              void* d_out, int out_size, void* d_ws, size_t ws_size,
                              hipStream_t stream) {
    const float* x  = (const float*)d_in[0];
    const int*   ei = (const int*)d_in[1];  // [2, E]
    const float* ea = (const float*)d_in[2];
    const float* W1 = (const float*)d_in[3];  // [129, 128]
    const float* b1 = (const float*)d_in[4];
    const float* W2 = (const float*)d_in[5];  // [257, 64]
    const float* b2 = (const float*)d_in[6];
    float* out = (float*)d_out;

    const int IN_C = 64;
    const int N = in_sizes[0] / IN_C;  // 50000 (multiple of 16)
    const int E = in_sizes[2];         // 800000
    const int* src = ei;
    const int* dst = ei + E;

    // workspace layout (floats): cnt[N] | s[N] | y[N*128] | h[N*128]
    float* cnt = (float*)d_ws;
    float* sEA = cnt + N;
    float* yAg = sEA + N;
    float* h   = yAg + (long long)N * 128;

    // zero cnt, s, y (contiguous)
    zero_f32<<<2048, 256, 0, stream>>>(cnt, (long long)N * (2 + 128));

    edge_stats<<<(E + 255) / 256, 256, 0, stream>>>(dst, ea, cnt, sEA, E);

    // layer 1: scatter x (64 ch) then fp32-WMMA GEMM 50000x128 @ 128x128 -> h
    long long t1 = (long long)E * 64;
    scatter_feat<6><<<(int)((t1 + 255) / 256), 256, 0, stream>>>(x, src, dst, yAg, E);
    mp_gemm<64, 128><<<N / 16, 256, 0, stream>>>(x, yAg, cnt, sEA, W1, b1, h);

    // layer 2: reset y, scatter h (128 ch), GEMM 50000x256 @ 256x64 -> out
    zero_f32<<<2048, 256, 0, stream>>>(yAg, (long long)N * 128);
    long long t2 = (long long)E * 128;
    scatter_feat<7><<<(int)((t2 + 255) / 256), 256, 0, stream>>>(h, src, dst, yAg, E);
    mp_gemm<128, 64><<<N / 16, 128, 0, stream>>>(h, yAg, cnt, sEA, W2, b2, out);

    log_softmax64<<<(N + 7) / 8, 256, 0, stream>>>(out, N);
}